// HWnet_plus_21251498180925
// MI455X (gfx1250) — compile-verified
//
#include <hip/hip_runtime.h>
#include <hip/hip_bf16.h>

typedef __attribute__((ext_vector_type(2))) float v2f;
typedef __attribute__((ext_vector_type(8))) float v8f;
typedef __attribute__((ext_vector_type(4))) float f4v;

#define T_BINS   16384
#define D_DIM    128
#define EDGE_SZ  4
#define W_WIN    9
#define SCPAD    12                  // padded score row stride (floats), 48B -> 16B aligned
#define TAKECARE 10.0f
#define BTILE    16                  // batch elements per block (WMMA M)
#define CHUNK_W  3                   // window slots per K-chunk
#define CROWS    (CHUNK_W * BTILE)   // 48 gathered rows per chunk
#define RPAD     144                 // padded LDS row stride (floats), conflict-free

__global__ __launch_bounds__(256, 1) void hwnet_wmma_kernel(
    const float* __restrict__ inp,
    const float* __restrict__ etab,
    const float* __restrict__ emin,
    const float* __restrict__ emax,
    const float* __restrict__ vtab,
    float* __restrict__ out,
    int Btot)
{
    __shared__ __align__(16) float sV[CROWS * RPAD];      // 27,648 B staged gather rows
    __shared__ __align__(16) float sScore[BTILE * SCPAD]; // softmax weights (padded)
    __shared__ int sBase[BTILE];                          // first row of each window

    const int tid = threadIdx.x;
    const int b0  = blockIdx.x * BTILE;

    // ---- stage 0: bin search + softmax scores (threads 0..15, one element each)
    if (tid < BTILE) {
        int b = b0 + tid;
        float x = (b < Btot) ? inp[b] : 0.0f;
        int j = (int)(x * (float)T_BINS);
        j = j < 0 ? 0 : j;
        j = j > T_BINS - 1 ? T_BINS - 1 : j;
        int guard = 0;
        while (j > 0 && x < emin[j] && guard < 8) { --j; ++guard; }
        guard = 0;
        while (j < T_BINS - 1 && x > emax[j] && guard < 8) { ++j; ++guard; }
        // reference argmax picks the FIRST bin containing x (shared-edge case)
        if (j > 0 && x >= emin[j - 1] && x <= emax[j - 1]) --j;

        float et   = etab[j];
        float wide = emax[j] - emin[j];
        float dist = (x - et) / wide;
        int jc = j < EDGE_SZ ? EDGE_SZ : j;
        jc = jc > T_BINS - 1 - EDGE_SZ ? T_BINS - 1 - EDGE_SZ : jc;
        int base = jc - EDGE_SZ;
        sBase[tid] = base;

        float e[W_WIN];
        float s = 0.0f;
        #pragma unroll
        for (int w = 0; w < W_WIN; ++w) {
            float off = (float)(jc - j + (w - EDGE_SZ));
            float d   = dist - off;
            float v   = __expf(-TAKECARE * d * d);
            e[w] = v; s += v;
            // warm L2 for the upcoming gather (CDNA5 global_prefetch_b8)
            __builtin_prefetch(vtab + (size_t)(base + w) * D_DIM, 0, 1);
        }
        float inv = 1.0f / s;
        #pragma unroll
        for (int w = 0; w < W_WIN; ++w)
            sScore[tid * SCPAD + w] = e[w] * inv;
    }
    __syncthreads();

    const int lane  = tid & 31;
    const int wav   = tid >> 5;        // n-tile: dims [wav*16, wav*16+16)
    const int mrow  = lane & 15;       // A: M row / B,D: N column (low half)
    const int khalf = lane >> 4;       // K split across half-waves
    const int ncol  = wav * 16 + mrow;
    const int tsel  = mrow - 2 * khalf;        // A slot matches when tsel == (4kk)&15 (+1)
    const int bbase = 2 * khalf * RPAD + ncol; // B-fragment LDS base (floats)

    // hoist this lane's 9 scores into registers (vector LDS loads, no per-step loads)
    f4v s03 = *(const f4v*)&sScore[mrow * SCPAD + 0];
    f4v s47 = *(const f4v*)&sScore[mrow * SCPAD + 4];
    float s8 = sScore[mrow * SCPAD + 8];
    const float sc[W_WIN] = { s03.x, s03.y, s03.z, s03.w, s47.x, s47.y, s47.z, s47.w, s8 };

    v8f acc = {0.f, 0.f, 0.f, 0.f, 0.f, 0.f, 0.f, 0.f};

    #pragma unroll
    for (int c = 0; c < 3; ++c) {
        // ---- cooperative coalesced gather: 48 rows x 128 f32 into LDS
        // 32 consecutive threads fetch one full 512B row (float4 each)
        #pragma unroll
        for (int i = 0; i < 6; ++i) {
            int flat = tid + 256 * i;      // < 1536 = 48 rows * 32 float4
            int row  = flat >> 5;          // chunk row id: wl*16 + m
            int col4 = flat & 31;
            int m    = row & 15;
            int wl   = row >> 4;
            int grow = sBase[m] + c * CHUNK_W + wl;
            f4v v = *(const f4v*)(vtab + (size_t)grow * D_DIM + col4 * 4);
            *(f4v*)&sV[row * RPAD + col4 * 4] = v;   // single ds_store_b128
        }
        __syncthreads();

        // ---- 12 K-steps of V_WMMA_F32_16X16X4_F32 over 48 block-diagonal rows
        // A[m,k] = score[m, c*3 + (k>>4)] iff (k&15)==m  =>  D = sum_w score_w * row_w
        #pragma unroll
        for (int kk = 0; kk < 12; ++kk) {
            const int   r  = (4 * kk) & 15;      // compile-time
            const float sw = sc[c * CHUNK_W + (kk >> 2)];  // compile-time index
            v2f a, bf;
            a.x = (tsel == r    ) ? sw : 0.0f;   // v_cmp + v_cndmask, no LDS, no exec games
            a.y = (tsel == r + 1) ? sw : 0.0f;
            bf.x = sV[bbase + kk * 4 * RPAD];          // immediate-offset ds_load_b32
            bf.y = sV[bbase + kk * 4 * RPAD + RPAD];
            acc = __builtin_amdgcn_wmma_f32_16x16x4_f32(false, a, false, bf,
                                                        (short)0, acc, false, false);
        }
        __syncthreads();
    }

    // ---- store D tile: VGPR v holds M = v + 8*(lane>>4), N = lane&15
    #pragma unroll
    for (int v = 0; v < 8; ++v) {
        int m = v + khalf * 8;
        int b = b0 + m;
        if (b < Btot)
            out[(size_t)b * D_DIM + ncol] = acc[v];
    }
}

extern "C" void kernel_launch(void* const* d_in, const int* in_sizes, int n_in,
                              void* d_out, int out_size, void* d_ws, size_t ws_size,
                              hipStream_t stream) {
    const float* inp  = (const float*)d_in[0]; // inputs [B,1]
    const float* etab = (const float*)d_in[1]; // evaluate_table [T,1]
    const float* emin = (const float*)d_in[2]; // evaluate_min_table [T,1]
    const float* emax = (const float*)d_in[3]; // evaluate_max_table [T,1]
    const float* vtab = (const float*)d_in[4]; // vector_table [T,D]
    float* out = (float*)d_out;

    int B = in_sizes[0];
    int nblk = (B + BTILE - 1) / BTILE;
    hipLaunchKernelGGL(hwnet_wmma_kernel, dim3(nblk), dim3(256), 0, stream,
                       inp, etab, emin, emax, vtab, out, B);
}